// SinkhornM_1425929142655
// MI455X (gfx1250) — compile-verified
//
#include <hip/hip_runtime.h>
#include <math.h>

typedef float v2f __attribute__((ext_vector_type(2)));
typedef float v8f __attribute__((ext_vector_type(8)));

#define LOWC  0.02f
#define HIGHC 0.98f
#define EPSC  1e-12f
#define LOG2E 1.44269504088896340736f

__device__ __forceinline__ v8f wmma_f32(v2f a, v2f b, v8f c) {
    // D = A(16x4,f32) * B(4x16,f32) + C(16x16,f32)
    return __builtin_amdgcn_wmma_f32_16x16x4_f32(
        /*neg_a=*/false, a, /*neg_b=*/false, b,
        /*c_mod=*/(short)0, c, /*reuse_a=*/false, /*reuse_b=*/false);
}

// Single-instruction TRANS-pipe math (co-executes with VALU on CDNA5).
__device__ __forceinline__ float fexp(float x)  { return __builtin_amdgcn_exp2f(x * LOG2E); }
__device__ __forceinline__ float frcp(float x)  { return __builtin_amdgcn_rcpf(x); }
__device__ __forceinline__ float fsqrt(float x) { return __builtin_amdgcn_sqrtf(x); }
__device__ __forceinline__ float sqsf(float x) {
    // LOW + (HIGH-LOW) * sigmoid(x)
    return LOWC + (HIGHC - LOWC) * frcp(1.0f + fexp(-x));
}

// Per-wave LDS layout (floats): h1[16*34] | h2[16*18] | pars[32*17]
#define H1_STRIDE 34
#define H2_STRIDE 18
#define PR_STRIDE 17
#define WAVE_LDS  (16*H1_STRIDE + 16*H2_STRIDE + 32*PR_STRIDE) // 544+288+544 = 1376

__global__ __launch_bounds__(256) void sinkhorn_mlp_kernel(
    const float* __restrict__ margins,
    const float* __restrict__ W1, const float* __restrict__ b1,
    const float* __restrict__ W2, const float* __restrict__ b2,
    const float* __restrict__ W3, const float* __restrict__ b3,
    float* __restrict__ out, int B)
{
    __shared__ float lds[8 * WAVE_LDS];

    const int lane = threadIdx.x & 31;
    const int wave = threadIdx.x >> 5;
    const int n    = lane & 15;   // row (A-frag) / col (B,C-frag) index
    const int hk   = lane >> 4;   // K half-select within a fragment

    float* h1  = lds + wave * WAVE_LDS;
    float* h2  = h1 + 16 * H1_STRIDE;
    float* prs = h2 + 16 * H2_STRIDE;

    const long waveRow0 = ((long)blockIdx.x * 8 + wave) * 32;

    // ---- Preload weight B-fragments (shared across both 16-row tiles) ----
    // B layout (4x16 f32): VGPR j, lane (n + 16*hk) holds B[2*hk + j][n]
    v2f w1f[2][2]; // [n-tile t][k-chunk c]
    #pragma unroll
    for (int t = 0; t < 2; ++t)
        #pragma unroll
        for (int c = 0; c < 2; ++c) {
            int k = c * 4 + hk * 2;
            w1f[t][c].x = W1[(k + 0) * 32 + t * 16 + n];
            w1f[t][c].y = W1[(k + 1) * 32 + t * 16 + n];
        }
    v2f w2f[8];
    #pragma unroll
    for (int c = 0; c < 8; ++c) {
        int k = c * 4 + hk * 2;
        w2f[c].x = W2[(k + 0) * 16 + n];
        w2f[c].y = W2[(k + 1) * 16 + n];
    }
    v2f w3f[4]; // W3 is 16x9, pad N to 16 with zeros
    #pragma unroll
    for (int c = 0; c < 4; ++c) {
        int k = c * 4 + hk * 2;
        w3f[c].x = (n < 9) ? W3[(k + 0) * 9 + n] : 0.0f;
        w3f[c].y = (n < 9) ? W3[(k + 1) * 9 + n] : 0.0f;
    }
    const float bias1_0 = b1[n], bias1_1 = b1[16 + n];
    const float bias2   = b2[n];
    const float bias3   = (n < 9) ? b3[n] : 0.0f;

    // ---- Two 16-row tiles through the MLP via f32 WMMA ----
    #pragma unroll
    for (int T = 0; T < 2; ++T) {
        long arow = waveRow0 + T * 16 + n;
        if (arow >= B) arow = B - 1;           // clamp (EXEC must stay all-1s for WMMA)
        const float* mp = margins + arow * 8;
        // A layout (16x4 f32): lane m=n, VGPR j holds A[m][2*hk + j]
        v2f a0, a1;
        a0.x = mp[hk * 2 + 0]; a0.y = mp[hk * 2 + 1];         // K chunk 0..3
        a1.x = mp[4 + hk * 2 + 0]; a1.y = mp[4 + hk * 2 + 1]; // K chunk 4..7

        // Layer 1: (16x8)@(8x32) -> two 16x16 C tiles
        v8f c10 = {0.f,0.f,0.f,0.f,0.f,0.f,0.f,0.f};
        v8f c11 = {0.f,0.f,0.f,0.f,0.f,0.f,0.f,0.f};
        c10 = wmma_f32(a0, w1f[0][0], c10);
        c10 = wmma_f32(a1, w1f[0][1], c10);
        c11 = wmma_f32(a0, w1f[1][0], c11);
        c11 = wmma_f32(a1, w1f[1][1], c11);
        // bias + relu, C-layout -> LDS row-major (stride 34)
        #pragma unroll
        for (int v = 0; v < 8; ++v) {
            int row = v + 8 * hk;
            h1[row * H1_STRIDE + n]      = fmaxf(c10[v] + bias1_0, 0.0f);
            h1[row * H1_STRIDE + 16 + n] = fmaxf(c11[v] + bias1_1, 0.0f);
        }

        // Layer 2: (16x32)@(32x16), 8 K-chunks, A re-read from LDS in A-layout
        v8f c2 = {0.f,0.f,0.f,0.f,0.f,0.f,0.f,0.f};
        #pragma unroll
        for (int c = 0; c < 8; ++c) {
            v2f a;
            a.x = h1[n * H1_STRIDE + c * 4 + hk * 2 + 0];
            a.y = h1[n * H1_STRIDE + c * 4 + hk * 2 + 1];
            c2 = wmma_f32(a, w2f[c], c2);
        }
        #pragma unroll
        for (int v = 0; v < 8; ++v) {
            int row = v + 8 * hk;
            h2[row * H2_STRIDE + n] = fmaxf(c2[v] + bias2, 0.0f);
        }

        // Layer 3: (16x16)@(16x9 padded to 16), 4 K-chunks
        v8f c3 = {0.f,0.f,0.f,0.f,0.f,0.f,0.f,0.f};
        #pragma unroll
        for (int c = 0; c < 4; ++c) {
            v2f a;
            a.x = h2[n * H2_STRIDE + c * 4 + hk * 2 + 0];
            a.y = h2[n * H2_STRIDE + c * 4 + hk * 2 + 1];
            c3 = wmma_f32(a, w3f[c], c3);
        }
        #pragma unroll
        for (int v = 0; v < 8; ++v) {
            int row = v + 8 * hk;
            prs[(T * 16 + row) * PR_STRIDE + n] = c3[v] + bias3;
        }
    }

    // Same-wave DS ops are in-order; fence so compiler doesn't hoist the reads.
    asm volatile("s_wait_dscnt 0" ::: "memory");

    // ---- Per-lane tail: one row per lane (32 rows per wave) ----
    const long g  = waveRow0 + lane;
    const bool valid = (g < (long)B);
    const long gc = valid ? g : (long)B - 1;

    float p[9];
    #pragma unroll
    for (int i = 0; i < 9; ++i) p[i] = prs[lane * PR_STRIDE + i];

    const float* mr = margins + gc * 8;
    const float M0 = mr[0], M1 = mr[1], M2 = mr[2];
    const float F0 = mr[3], F1 = mr[4], F2 = mr[5];

    // K = tau_batched(exp(pars[0:4]))  (the fixed kernel matrix of the scaling)
    const float K00 = fexp(p[0]), K01 = fexp(p[1]);
    const float K10 = fexp(p[2]), K11 = fexp(p[3]);
    const float K02 = fsqrt(K00 * K01), K12 = fsqrt(K10 * K11);
    const float K20 = fsqrt(K00 * K10), K21 = fsqrt(K01 * K11);
    const float K22 = fsqrt(K20 * K21);

    const float s4 = sqsf(p[4]), s5 = sqsf(p[5]);
    const float s6 = sqsf(p[6]), s7 = sqsf(p[7]);
    const float V  = fexp(p[8]);

    const float r0 = M0 * s4, r1 = M1 * s5, r2 = M2;          // row margins (shm0[2]=1)
    const float c0 = F0 * s6, c1 = F1 * s7, c2 = F2;          // col margins
    const float mum0_0 = M0 * (1.0f - s4), mum0_1 = M1 * (1.0f - s5), mum0_2 = M2 * 0.0f;
    const float mu0f_0 = F0 * (1.0f - s6), mu0f_1 = F1 * (1.0f - s7), mu0f_2 = F2 * 0.0f;

    // 10 Sinkhorn iterations in u/v scaling-vector form:
    //   A stays diag(u)*K*diag(v);  u = r/(K v + EPS),  v = c/(K^T u + EPS).
    // EPS folded into the FMA-chain seed; rcp on the TRANS pipe. Values are
    // O(0.01..1) vs EPS=1e-12, so this matches the reference to fp32 noise.
    float u0, u1, u2;
    float v0 = 1.0f, v1 = 1.0f, v2 = 1.0f;
    #pragma unroll
    for (int it = 0; it < 10; ++it) {
        float kv0 = fmaf(K02, v2, fmaf(K01, v1, fmaf(K00, v0, EPSC)));
        float kv1 = fmaf(K12, v2, fmaf(K11, v1, fmaf(K10, v0, EPSC)));
        float kv2 = fmaf(K22, v2, fmaf(K21, v1, fmaf(K20, v0, EPSC)));
        u0 = r0 * frcp(kv0);
        u1 = r1 * frcp(kv1);
        u2 = r2 * frcp(kv2);
        float ku0 = fmaf(K20, u2, fmaf(K10, u1, fmaf(K00, u0, EPSC)));
        float ku1 = fmaf(K21, u2, fmaf(K11, u1, fmaf(K01, u0, EPSC)));
        float ku2 = fmaf(K22, u2, fmaf(K12, u1, fmaf(K02, u0, EPSC)));
        v0 = c0 * frcp(ku0);
        v1 = c1 * frcp(ku1);
        v2 = c2 * frcp(ku2);
    }

    if (valid) {
        float4* o = (float4*)(out + g * 16);
        o[0] = float4{(K00 * u0) * v0, (K01 * u0) * v1, (K02 * u0) * v2, mum0_0};
        o[1] = float4{(K10 * u1) * v0, (K11 * u1) * v1, (K12 * u1) * v2, mum0_1};
        o[2] = float4{(K20 * u2) * v0, (K21 * u2) * v1, (K22 * u2) * v2, mum0_2};
        o[3] = float4{mu0f_0, mu0f_1, mu0f_2, 0.0f};
        out[(long)B * 16 + g] = V;
    }
}

extern "C" void kernel_launch(void* const* d_in, const int* in_sizes, int n_in,
                              void* d_out, int out_size, void* d_ws, size_t ws_size,
                              hipStream_t stream) {
    const float* margins = (const float*)d_in[0];
    const float* W1 = (const float*)d_in[1];
    const float* b1 = (const float*)d_in[2];
    const float* W2 = (const float*)d_in[3];
    const float* b2 = (const float*)d_in[4];
    const float* W3 = (const float*)d_in[5];
    const float* b3 = (const float*)d_in[6];
    float* out = (float*)d_out;

    const int B = in_sizes[0] / 8;               // margins is (B, 8)
    const int rowsPerBlock = 256;                // 8 waves * 32 rows
    const int grid = (B + rowsPerBlock - 1) / rowsPerBlock;
    sinkhorn_mlp_kernel<<<grid, 256, 0, stream>>>(margins, W1, b1, W2, b2, W3, b3, out, B);
}